// HeteroConv_52570399703510
// MI455X (gfx1250) — compile-verified
//
#include <hip/hip_runtime.h>

#define DD 128
#define EPSV 1e-5f

typedef float v2f __attribute__((ext_vector_type(2)));
typedef float v8f __attribute__((ext_vector_type(8)));

// ---------------- utility kernels ----------------

__global__ __launch_bounds__(256) void k_zero(float* __restrict__ p, long n) {
  long i = (long)blockIdx.x * blockDim.x + threadIdx.x;
  if (i < n) p[i] = 0.0f;
}

// deg[k][node] += 1 for every edge (dst laid out [3][E])
__global__ __launch_bounds__(256) void k_count_deg(const int* __restrict__ dst,
                                                   float* __restrict__ deg,
                                                   int nedges, int nnodes) {
  int t = blockIdx.x * blockDim.x + threadIdx.x;
  if (t >= nedges * 3) return;
  int k = t / nedges;
  atomicAdd(&deg[(long)k * nnodes + dst[t]], 1.0f);
}

__global__ __launch_bounds__(256) void k_invdeg(float* __restrict__ deg, long n) {
  long i = (long)blockIdx.x * blockDim.x + threadIdx.x;
  if (i < n) deg[i] = 1.0f / fmaxf(deg[i], 1.0f);
}

// one wave per edge: lane handles 4 consecutive floats of the 128-float row
__global__ __launch_bounds__(256) void k_scatter(const float* __restrict__ h,
                                                 const int* __restrict__ src,
                                                 const int* __restrict__ dst,
                                                 float* __restrict__ neigh,
                                                 int nedges) {
  int t = blockIdx.x * blockDim.x + threadIdx.x;
  int e = t >> 5;
  if (e >= nedges) return;
  int c = (t & 31) << 2;
  int s = src[e];
  int d = dst[e];
  const float4 v = *(const float4*)(h + (long)s * DD + c);
  float* p = neigh + (long)d * DD + c;
  atomicAdd(p + 0, v.x);
  atomicAdd(p + 1, v.y);
  atomicAdd(p + 2, v.z);
  atomicAdd(p + 3, v.w);
}

// ---------------- fp32 WMMA GEMM ----------------
// One wave computes a 16x128 tile of (A @ W) accumulated into c[8] (v8f each).
// A fragment (16x4 f32): lane m (0..15) = row m; half=lane>>4 selects K {0,1} vs {2,3}.
// B fragment (4x16 f32): lane m = column m; VGPR j = K row (2*half + j).
__device__ __forceinline__ void wmma_KxD(const float* __restrict__ A, float ascale,
                                         const float* __restrict__ W,
                                         int row, int m, int half, v8f c[8]) {
  for (int kt = 0; kt < DD; kt += 4) {
    v2f a = *(const v2f*)(A + (long)row * DD + kt + 2 * half);
    a[0] *= ascale;
    a[1] *= ascale;
    const float* Wk = W + (long)(kt + 2 * half) * DD + m;
#pragma unroll
    for (int nt = 0; nt < 8; ++nt) {
      v2f bf;
      bf[0] = Wk[nt * 16];        // K row (kt + 2*half + 0)
      bf[1] = Wk[DD + nt * 16];   // K row (kt + 2*half + 1)
      c[nt] = __builtin_amdgcn_wmma_f32_16x16x4_f32(
          false, a, false, bf, (short)0, c[nt], false, false);
    }
  }
}

// acc[tile] (init? = : +=) relu?( h@Ws + (neigh*invdeg)@Wn + bias )
__global__ __launch_bounds__(256) void k_gemm(const float* __restrict__ h,
                                              const float* __restrict__ neigh,
                                              const float* __restrict__ invdeg,
                                              const float* __restrict__ Ws,
                                              const float* __restrict__ Wn,
                                              const float* __restrict__ bias,
                                              float* __restrict__ acc,
                                              int nrows, int initf, int reluf) {
  int wave = threadIdx.x >> 5;
  int lane = threadIdx.x & 31;
  int row0 = (blockIdx.x * 8 + wave) * 16;
  if (row0 >= nrows) return;          // N is a multiple of 16; tiles are full or empty
  int m = lane & 15;
  int half = lane >> 4;
  int row = row0 + m;

  v8f c[8];
#pragma unroll
  for (int i = 0; i < 8; ++i) c[i] = (v8f){0.f, 0.f, 0.f, 0.f, 0.f, 0.f, 0.f, 0.f};

  wmma_KxD(h, 1.0f, Ws, row, m, half, c);
  wmma_KxD(neigh, invdeg[row], Wn, row, m, half, c);

  // C/D layout: VGPR r -> M = r (lanes 0..15) or r+8 (lanes 16..31); N = nt*16 + m
#pragma unroll
  for (int nt = 0; nt < 8; ++nt) {
    float bv = bias[nt * 16 + m];
#pragma unroll
    for (int r = 0; r < 8; ++r) {
      float v = c[nt][r] + bv;
      if (reluf) v = fmaxf(v, 0.0f);
      long o = (long)(row0 + r + 8 * half) * DD + nt * 16 + m;
      if (initf) acc[o] = v;
      else       acc[o] += v;
    }
  }
}

// ---------------- BatchNorm ----------------

__global__ __launch_bounds__(DD) void k_bnstats(const float* __restrict__ acc,
                                                float* __restrict__ stats,
                                                int nrows, int rows_per_block) {
  int j = threadIdx.x;                       // column
  int r0 = blockIdx.x * rows_per_block;
  int r1 = min(r0 + rows_per_block, nrows);
  float s = 0.0f, q = 0.0f;
  for (int r = r0; r < r1; ++r) {
    float v = acc[(long)r * DD + j];
    s += v;
    q += v * v;
  }
  atomicAdd(&stats[j], s);
  atomicAdd(&stats[DD + j], q);
}

__global__ __launch_bounds__(256) void k_bnapply(const float* __restrict__ acc,
                                                 const float* __restrict__ stats,
                                                 const float* __restrict__ gamma,
                                                 const float* __restrict__ beta,
                                                 float* __restrict__ out,
                                                 long n, float invN) {
  long i = (long)blockIdx.x * blockDim.x + threadIdx.x;
  if (i >= n) return;
  int j = (int)(i & (DD - 1));
  float mu = stats[j] * invN;
  float var = stats[DD + j] * invN - mu * mu;
  float v = (acc[i] - mu) * rsqrtf(var + EPSV);
  out[i] = gamma[j] * v + beta[j];
}

// ---------------- driver ----------------

extern "C" void kernel_launch(void* const* d_in, const int* in_sizes, int n_in,
                              void* d_out, int out_size, void* d_ws, size_t ws_size,
                              hipStream_t stream) {
  (void)n_in; (void)out_size; (void)ws_size;
  const float* x      = (const float*)d_in[0];
  const int*   src    = (const int*)d_in[1];
  const int*   dst    = (const int*)d_in[2];
  const float* W_self = (const float*)d_in[3];
  const float* W_neigh= (const float*)d_in[4];
  const float* bvec   = (const float*)d_in[5];
  const float* gamma  = (const float*)d_in[6];
  const float* beta   = (const float*)d_in[7];

  const int N = in_sizes[0] / DD;     // 100000
  const int E = in_sizes[1] / 3;      // 600000
  const long ND = (long)N * DD;

  // workspace layout
  float* h_cur  = (float*)d_ws;       // N*D
  float* accb   = h_cur + ND;         // N*D
  float* neigh  = accb + ND;          // N*D
  float* invdeg = neigh + ND;         // 3*N
  float* stats  = invdeg + 3L * N;    // 2*D

  // per-etype inverse degrees (reused across layers)
  k_zero<<<(int)((3L * N + 255) / 256), 256, 0, stream>>>(invdeg, 3L * N);
  k_count_deg<<<(E * 3 + 255) / 256, 256, 0, stream>>>(dst, invdeg, E, N);
  k_invdeg<<<(int)((3L * N + 255) / 256), 256, 0, stream>>>(invdeg, 3L * N);

  hipMemcpyAsync(h_cur, x, ND * sizeof(float), hipMemcpyDeviceToDevice, stream);

  const int gemmBlocks = (N + 127) / 128;               // 8 waves x 16 rows per block
  const int scatBlocks = (int)(((long)E * 32 + 255) / 256);
  const int zeroBlocks = (int)((ND + 255) / 256);
  const int rpb = 256;
  const int statBlocks = (N + rpb - 1) / rpb;

  for (int l = 0; l < 3; ++l) {
    for (int k = 0; k < 3; ++k) {
      k_zero<<<zeroBlocks, 256, 0, stream>>>(neigh, ND);
      k_scatter<<<scatBlocks, 256, 0, stream>>>(h_cur, src + (long)k * E,
                                                dst + (long)k * E, neigh, E);
      k_gemm<<<gemmBlocks, 256, 0, stream>>>(
          h_cur, neigh, invdeg + (long)k * N,
          W_self  + (long)(l * 3 + k) * DD * DD,
          W_neigh + (long)(l * 3 + k) * DD * DD,
          bvec    + (long)(l * 3 + k) * DD,
          accb, N, (k == 0) ? 1 : 0, (l < 2) ? 1 : 0);
    }
    k_zero<<<1, 256, 0, stream>>>(stats, 2 * DD);
    k_bnstats<<<statBlocks, DD, 0, stream>>>(accb, stats, N, rpb);
    float* outp = (l == 2) ? (float*)d_out : h_cur;
    k_bnapply<<<zeroBlocks, 256, 0, stream>>>(accb, stats, gamma + l * DD,
                                              beta + l * DD, outp, ND, 1.0f / N);
  }
}